// CompressiveMemory_45561013076529
// MI455X (gfx1250) — compile-verified
//
#include <hip/hip_runtime.h>

typedef __attribute__((ext_vector_type(2))) float v2f;
typedef __attribute__((ext_vector_type(8))) float v8f;

#define NB 4
#define NH 16
#define NS 4096
#define ND 64
#define CH 8                 // S-chunks per (b,h)
#define SCH (NS / CH)        // 512 rows per chunk
#define SS  128              // rows per superstep (8 waves x 16 rows)
#define NSS (SCH / SS)       // 4 supersteps per chunk
#define BH  (NB * NH)        // 64

#define MEM_STRIDE 65        // 64x64 mem matrix, padded
#define SK_STRIDE  66        // staged sigma_k / value_delta tiles, padded (keeps float2 8B-aligned)

__global__ __launch_bounds__(256) void cm_main(
    const float* __restrict__ q, const float* __restrict__ kmat,
    const float* __restrict__ v, const float* __restrict__ ao,
    const float* __restrict__ betas, const float* __restrict__ init_mem,
    const float* __restrict__ init_z,
    float* __restrict__ out, float* __restrict__ spart, float* __restrict__ zpart)
{
  __shared__ float mem_lds[ND * MEM_STRIDE];
  __shared__ float sk_lds[SS * SK_STRIDE];
  __shared__ float vd_lds[SS * SK_STRIDE];
  __shared__ float gate_lds[ND];
  __shared__ float z_lds[ND];

  const int blk  = blockIdx.x;
  const int ch   = blk & (CH - 1);
  const int bh   = blk / CH;
  const int h    = bh & (NH - 1);
  const int tid  = threadIdx.x;
  const int lane = tid & 31;
  const int wave = tid >> 5;

  // Stage per-head constants: memory matrix, z, gate = sigmoid(betas)
  const float* memp = init_mem + (size_t)h * ND * ND;
  for (int i = tid; i < ND * ND; i += 256)
    mem_lds[(i >> 6) * MEM_STRIDE + (i & 63)] = memp[i];
  if (tid < ND) {
    z_lds[tid] = init_z[h * ND + tid];
    gate_lds[tid] = 1.0f / (1.0f + __expf(-betas[h * ND + tid]));
  }
  __syncthreads();

  // WMMA f32 16x16x4 fragment coords: r = M (row), kb = K-pair base per lane half
  const int r  = lane & 15;
  const int hi = lane >> 4;
  const int kb = hi << 1;

  float zreg[32];
  #pragma unroll
  for (int kk = 0; kk < 16; ++kk) {
    zreg[2 * kk]     = z_lds[4 * kk + kb];
    zreg[2 * kk + 1] = z_lds[4 * kk + kb + 1];
  }
  float g4[4];
  #pragma unroll
  for (int nt = 0; nt < 4; ++nt) g4[nt] = gate_lds[nt * 16 + r];

  // State accumulation: wave owns tiles (mt, nt0) and (mt, nt0+1) of the 64x64 state
  const int mt  = wave >> 1;
  const int nt0 = (wave & 1) << 1;
  v8f sacc0 = {};
  v8f sacc1 = {};
  float zacc = 0.0f;
  const int zj = tid & 63;
  const int zg = tid >> 6;

  const size_t rowbase = ((size_t)bh * NS + (size_t)ch * SCH) * ND;

  for (int ss = 0; ss < NSS; ++ss) {
    __syncthreads();  // previous superstep's LDS consumers done before restaging
    const int s_tile = ss * SS + wave * 16;
    const float* qrow = q    + rowbase + (size_t)(s_tile + r) * ND;
    const float* krow = kmat + rowbase + (size_t)(s_tile + r) * ND;

    // ---- sigma_q in A-fragment layout + denominator dot(sigma_q, z) ----
    v2f sq[16];
    float dq = 0.0f;
    #pragma unroll
    for (int kk = 0; kk < 16; ++kk) {
      v2f t = *(const v2f*)(qrow + 4 * kk + kb);
      t.x = (t.x > 0.0f) ? (t.x + 1.0f) : __expf(t.x);
      t.y = (t.y > 0.0f) ? (t.y + 1.0f) : __expf(t.y);
      sq[kk] = t;
      dq += t.x * zreg[2 * kk] + t.y * zreg[2 * kk + 1];
    }
    dq += __shfl_xor(dq, 16, 32);
    const float rq = 1.0f / dq;
    float rqrow[8];
    #pragma unroll
    for (int j = 0; j < 8; ++j) rqrow[j] = __shfl(rq, 8 * hi + j, 32);

    // ---- memory_output = (sigma_q @ M) * (1/denom); blend with attention ----
    #pragma unroll
    for (int nt = 0; nt < 4; ++nt) {
      v8f acc = {};
      #pragma unroll
      for (int kk = 0; kk < 16; ++kk) {
        v2f bf;
        bf.x = mem_lds[(4 * kk + kb) * MEM_STRIDE + nt * 16 + r];
        bf.y = mem_lds[(4 * kk + kb + 1) * MEM_STRIDE + nt * 16 + r];
        acc = __builtin_amdgcn_wmma_f32_16x16x4_f32(false, sq[kk], false, bf,
                                                    (short)0, acc, false, false);
      }
      #pragma unroll
      for (int j = 0; j < 8; ++j) {
        const size_t idx = rowbase + (size_t)(s_tile + 8 * hi + j) * ND + nt * 16 + r;
        out[idx] = g4[nt] * acc[j] * rqrow[j] + (1.0f - g4[nt]) * ao[idx];
      }
    }

    // ---- sigma_k + its denominator ----
    v2f sk[16];
    float dk = 0.0f;
    #pragma unroll
    for (int kk = 0; kk < 16; ++kk) {
      v2f t = *(const v2f*)(krow + 4 * kk + kb);
      t.x = (t.x > 0.0f) ? (t.x + 1.0f) : __expf(t.x);
      t.y = (t.y > 0.0f) ? (t.y + 1.0f) : __expf(t.y);
      sk[kk] = t;
      dk += t.x * zreg[2 * kk] + t.y * zreg[2 * kk + 1];
    }
    dk += __shfl_xor(dk, 16, 32);
    const float rk = 1.0f / dk;
    float rkrow[8];
    #pragma unroll
    for (int j = 0; j < 8; ++j) rkrow[j] = __shfl(rk, 8 * hi + j, 32);

    // Stage sigma_k tile row-major into LDS
    const int sl = wave * 16 + r;
    #pragma unroll
    for (int kk = 0; kk < 16; ++kk)
      *(v2f*)&sk_lds[sl * SK_STRIDE + 4 * kk + kb] = sk[kk];

    // ---- retrieved = (sigma_k @ M)/denom ; stage value_delta = v - retrieved ----
    #pragma unroll
    for (int nt = 0; nt < 4; ++nt) {
      v8f acc = {};
      #pragma unroll
      for (int kk = 0; kk < 16; ++kk) {
        v2f bf;
        bf.x = mem_lds[(4 * kk + kb) * MEM_STRIDE + nt * 16 + r];
        bf.y = mem_lds[(4 * kk + kb + 1) * MEM_STRIDE + nt * 16 + r];
        acc = __builtin_amdgcn_wmma_f32_16x16x4_f32(false, sk[kk], false, bf,
                                                    (short)0, acc, false, false);
      }
      #pragma unroll
      for (int j = 0; j < 8; ++j) {
        const size_t idx = rowbase + (size_t)(s_tile + 8 * hi + j) * ND + nt * 16 + r;
        vd_lds[(wave * 16 + 8 * hi + j) * SK_STRIDE + nt * 16 + r] =
            v[idx] - acc[j] * rkrow[j];
      }
    }

    __syncthreads();  // tiles staged; begin cross-wave consumption

    // ---- new_z partial column sums (deterministic per-thread accumulation) ----
    for (int s = zg; s < SS; s += 4) zacc += sk_lds[s * SK_STRIDE + zj];

    // ---- new_states partial: sigma_k^T @ value_delta via WMMA, K = 128 rows ----
    #pragma unroll 4
    for (int ks = 0; ks < SS / 4; ++ks) {
      const int s0 = ks * 4;
      v2f a, b0, b1;
      a.x  = sk_lds[(s0 + kb) * SK_STRIDE + mt * 16 + r];       // A = sigma_k^T (M=d, K=s)
      a.y  = sk_lds[(s0 + kb + 1) * SK_STRIDE + mt * 16 + r];
      b0.x = vd_lds[(s0 + kb) * SK_STRIDE + nt0 * 16 + r];      // B = value_delta (K=s, N=d2)
      b0.y = vd_lds[(s0 + kb + 1) * SK_STRIDE + nt0 * 16 + r];
      b1.x = vd_lds[(s0 + kb) * SK_STRIDE + (nt0 + 1) * 16 + r];
      b1.y = vd_lds[(s0 + kb + 1) * SK_STRIDE + (nt0 + 1) * 16 + r];
      sacc0 = __builtin_amdgcn_wmma_f32_16x16x4_f32(false, a, false, b0, (short)0, sacc0, false, false);
      sacc1 = __builtin_amdgcn_wmma_f32_16x16x4_f32(false, a, false, b1, (short)0, sacc1, false, false);
    }
  }

  // ---- write per-chunk partials to workspace ----
  float* sp = spart + ((size_t)bh * CH + ch) * (ND * ND);
  #pragma unroll
  for (int j = 0; j < 8; ++j) {
    const int rowp = mt * 16 + 8 * hi + j;
    sp[rowp * ND + nt0 * 16 + r]       = sacc0[j];
    sp[rowp * ND + (nt0 + 1) * 16 + r] = sacc1[j];
  }
  float* zp = zpart + ((size_t)bh * CH + ch) * 256;
  zp[zg * 64 + zj] = zacc;
}

__global__ __launch_bounds__(256) void cm_reduce(
    const float* __restrict__ init_mem, const float* __restrict__ init_z,
    const float* __restrict__ spart, const float* __restrict__ zpart,
    float* __restrict__ out_states, float* __restrict__ out_z)
{
  const int bh = blockIdx.x;
  const int h  = bh & (NH - 1);
  const int t  = threadIdx.x;
  for (int e = t; e < ND * ND; e += 256) {
    float s = init_mem[(size_t)h * ND * ND + e];
    #pragma unroll
    for (int ch = 0; ch < CH; ++ch)
      s += spart[((size_t)bh * CH + ch) * (ND * ND) + e];
    out_states[(size_t)bh * ND * ND + e] = s;
  }
  if (t < ND) {
    float s = init_z[h * ND + t];
    #pragma unroll
    for (int ch = 0; ch < CH; ++ch)
      #pragma unroll
      for (int g = 0; g < 4; ++g)
        s += zpart[((size_t)bh * CH + ch) * 256 + g * 64 + t];
    out_z[(size_t)bh * ND + t] = s;
  }
}

extern "C" void kernel_launch(void* const* d_in, const int* in_sizes, int n_in,
                              void* d_out, int out_size, void* d_ws, size_t ws_size,
                              hipStream_t stream) {
  (void)in_sizes; (void)n_in; (void)out_size; (void)ws_size;
  const float* q        = (const float*)d_in[0];
  const float* k        = (const float*)d_in[1];
  const float* v        = (const float*)d_in[2];
  const float* ao       = (const float*)d_in[3];
  const float* betas    = (const float*)d_in[4];
  const float* init_mem = (const float*)d_in[5];
  const float* init_z   = (const float*)d_in[6];

  float* out        = (float*)d_out;
  float* out_states = out + (size_t)NB * NH * NS * ND;        // after [B,H,S,D]
  float* out_z      = out_states + (size_t)NB * NH * ND * ND; // after [B,H,D,D]

  float* spart = (float*)d_ws;                                // [BH][CH][64*64]
  float* zpart = spart + (size_t)BH * CH * ND * ND;           // [BH][CH][4][64]

  cm_main<<<BH * CH, 256, 0, stream>>>(q, k, v, ao, betas, init_mem, init_z,
                                       out, spart, zpart);
  cm_reduce<<<BH, 256, 0, stream>>>(init_mem, init_z, spart, zpart,
                                    out_states, out_z);
}